// StreamingKVCache_43069932045033
// MI455X (gfx1250) — compile-verified
//
#include <hip/hip_runtime.h>
#include <hip/hip_bf16.h>
#include <stdint.h>

// Streaming KV cache update for MI455X (gfx1250, wave32).
// Pure data movement: ~544 MB total traffic -> HBM-bound (~23us at 23.3 TB/s).
// One 256-thread block per (b, l) row. K plane: direct NT b128 copy through
// VGPRs. V plane: CDNA5 async copy global -> LDS -> global (ASYNCcnt path),
// per-lane disjoint 16B LDS slots, same-wave ordering via s_wait_asynccnt.

typedef float v4f __attribute__((ext_vector_type(4)));

#define PAGE_F4_PER_SLOT 256   /* H*D=1024 floats = 256 float4 per (slot, kv) */
#define PAGE_F4_PER_KV  4096   /* 16 slots * 256 f4 per kv plane of a page */
#define PAGE_F4         8192   /* 2 kv planes per page */
#define BUDGET          4096
#define SINK               4

__global__ __launch_bounds__(256) void StreamingKVCache_43069932045033_kernel(
    const float* __restrict__ cache,   // (P, 2, 16, H, D) fp32
    const float* __restrict__ knew,    // (B, S, H, D) fp32
    const float* __restrict__ vnew,    // (B, S, H, D) fp32
    const int*   __restrict__ qlens,   // (B,) int32
    const int*   __restrict__ ctxp,    // scalar
    const int*   __restrict__ slenp,   // scalar
    float*       __restrict__ out)     // (P, 2, 16, H, D) fp32
{
    const int l   = (int)blockIdx.x;          // cache position within batch
    const int b   = (int)blockIdx.y;          // batch
    const int L   = (int)gridDim.x;           // 4096
    const int tid = (int)threadIdx.x;         // 0..255

    // Uniform scalars (per-block) -> scalar loads
    const int ql   = qlens[b];
    const int ctx  = *ctxp;
    const int slen = *slenp;

    // ---- source selection (reference mask precedence: plain > shift > append > copy)
    int mode = 0;            // 0: gather from cache at s ; 1: gather from new k/v at s
    int s    = l;            // default: straight copy
    if (ql > 0) {
        if (ctx + ql <= BUDGET) {
            if (l >= ctx && l < ctx + ql) {                 // plain append
                mode = 1;
                s = l - ctx;
                s = s < 0 ? 0 : (s > slen - 1 ? slen - 1 : s);
            }
        } else {
            const int ub = ctx < BUDGET ? ctx : BUDGET;
            if (l >= SINK && l < BUDGET - ql) {             // rolling shift
                s = l + ql + (ub - BUDGET);
                s = s < 0 ? 0 : (s > L - 1 ? L - 1 : s);
            } else if (l >= BUDGET - ql && l < BUDGET) {    // rolling append
                mode = 1;
                s = l - (BUDGET - ql);
                s = s < 0 ? 0 : (s > slen - 1 ? slen - 1 : s);
            }
        }
    }

    // ---- destination offset (float4 units) in paged (P,2,16,H,D) layout
    const long q    = (long)b * L + l;
    const long dstK = (q >> 4) * PAGE_F4 + (q & 15) * PAGE_F4_PER_SLOT;  // kv=0

    // ---- source base pointers (float4 units)
    const v4f* srcK;
    const v4f* srcV;
    if (mode == 1) {
        const long n4 = ((long)b * slen + s) * PAGE_F4_PER_SLOT;
        srcK = (const v4f*)knew + n4;
        srcV = (const v4f*)vnew + n4;
    } else {
        const long q2 = (long)b * L + s;
        const long c4 = (q2 >> 4) * PAGE_F4 + (q2 & 15) * PAGE_F4_PER_SLOT;
        srcK = (const v4f*)cache + c4;
        srcV = srcK + PAGE_F4_PER_KV;
    }

    // ---- V plane: async global -> LDS (per-lane 16B slot), ASYNCcnt-tracked
    __shared__ __align__(16) unsigned char smem[256 * 16];
    // Hardware maps flat shared addresses to LDS via addr[31:0] truncation
    // (ISA sec.10.2), so the low 32 bits of the generic pointer ARE the LDS addr.
    unsigned ldsOff = (unsigned)(uintptr_t)&smem[tid * 16];
    unsigned long long gsrcV = (unsigned long long)(uintptr_t)(srcV + tid);
    asm volatile("global_load_async_to_lds_b128 %0, %1, off"
                 :: "v"(ldsOff), "v"(gsrcV) : "memory");

    // ---- K plane: direct NT b128 copy (overlaps with async V load latency)
    v4f kk = __builtin_nontemporal_load(srcK + tid);
    v4f* dK = (v4f*)out + dstK + tid;
    __builtin_nontemporal_store(kk, dK);

    // ---- V plane: wait for LDS fill (same-wave ordering), then async LDS -> global
    asm volatile("s_wait_asynccnt 0x0" ::: "memory");
    unsigned long long gdstV = (unsigned long long)(uintptr_t)(dK + PAGE_F4_PER_KV);
    asm volatile("global_store_async_from_lds_b128 %0, %1, off"
                 :: "v"(gdstV), "v"(ldsOff) : "memory");
    // s_endpgm performs an implicit wait-idle, covering outstanding async stores.
}

extern "C" void kernel_launch(void* const* d_in, const int* in_sizes, int n_in,
                              void* d_out, int out_size, void* d_ws, size_t ws_size,
                              hipStream_t stream) {
    (void)n_in; (void)out_size; (void)d_ws; (void)ws_size;
    const float* cache = (const float*)d_in[0];
    const float* knew  = (const float*)d_in[1];
    const float* vnew  = (const float*)d_in[2];
    const int*   qlens = (const int*)d_in[3];
    const int*   ctxp  = (const int*)d_in[4];
    const int*   slenp = (const int*)d_in[5];
    float*       out   = (float*)d_out;

    const int  B = in_sizes[3];                                  // 8
    const long L = (long)in_sizes[0] / ((long)B * 2 * 1024);     // 4096

    dim3 grid((unsigned)L, (unsigned)B), block(256);
    StreamingKVCache_43069932045033_kernel<<<grid, block, 0, stream>>>(
        cache, knew, vnew, qlens, ctxp, slenp, out);
}